// HeteroSAGEEncoder_88648124991297
// MI455X (gfx1250) — compile-verified
//
#include <hip/hip_runtime.h>

#define FEAT_D 128
#define WC_PITCH 260    // [self 128 | neigh 128 | pad 4]; bank = (4n + k) % 64 both halves
#define OUT_PITCH 132   // output staging pitch (conflict-free row writes)

typedef __attribute__((ext_vector_type(2))) float v2f;
typedef __attribute__((ext_vector_type(8))) float v8f;

// ---------------------------------------------------------------- zero fill
__global__ void zero4_kernel(float4* __restrict__ p, long n4) {
  long i = (long)blockIdx.x * blockDim.x + threadIdx.x;
  long stride = (long)gridDim.x * blockDim.x;
  float4 z = make_float4(0.f, 0.f, 0.f, 0.f);
  for (; i < n4; i += stride) p[i] = z;
}

// ---------------------------------------------------------------- degrees
__global__ void degree_kernel(const int* __restrict__ e_src, const int* __restrict__ e_dst,
                              unsigned* __restrict__ deg_src, unsigned* __restrict__ deg_dst,
                              int E) {
  int i = blockIdx.x * blockDim.x + threadIdx.x;
  if (i < E) {
    atomicAdd(&deg_src[e_src[i]], 1u);
    atomicAdd(&deg_dst[e_dst[i]], 1u);
  }
}

__global__ void invdeg_kernel(const unsigned* __restrict__ deg, float* __restrict__ inv, int n) {
  int i = blockIdx.x * blockDim.x + threadIdx.x;
  if (i < n) {
    unsigned d = deg[i];
    inv[i] = 1.0f / (float)(d ? d : 1u);   // isolated nodes: acc row is 0 anyway
  }
}

// ------------------------------------------------- edge scatter (wave32 per edge)
// Feature matrices (51.2 MB each) fit in the 192 MB L2, so the random row gather
// and the f32 atomics are L2-resident. Each wave gathers one 512B row (float4 per
// lane, fully coalesced) and scatters with native global_atomic_add_f32.
__global__ void scatter_kernel(const float* __restrict__ x, const int* __restrict__ e_from,
                               const int* __restrict__ e_to, float* __restrict__ acc, int E) {
  int lane  = threadIdx.x & 31;
  int wave  = blockIdx.x * (blockDim.x >> 5) + (threadIdx.x >> 5);
  int nwave = gridDim.x * (blockDim.x >> 5);
  for (int e = wave; e < E; e += nwave) {
    int s = e_from[e];                 // wave-uniform -> scalar load
    int t = e_to[e];
    const float4 v = *(const float4*)(x + (long)s * FEAT_D + lane * 4);
    float* d = acc + (long)t * FEAT_D + lane * 4;
    unsafeAtomicAdd(d + 0, v.x);
    unsafeAtomicAdd(d + 1, v.y);
    unsafeAtomicAdd(d + 2, v.z);
    unsafeAtomicAdd(d + 3, v.w);
  }
}

// ------------------------------------------------- fused SAGE projection (WMMA f32)
// h[row,:] = relu( h[row,:] @ Wself^T + (acc[row,:]*inv_deg[row]) @ Wneigh^T + b )
// In-place is safe: each output row depends only on the same row of h and acc.
// Block = 256 threads (8 waves); block tile = 128 rows x 128 cols.
// Wave w -> rows [16w, 16w+16); 8 n-tiles of 16 cols -> 8 v8f accumulators.
// V_WMMA_F32_16X16X4_F32 fragment layouts (ISA 7.12.2):
//   A 16x4 : lane m=L&15; lanes<16 hold K{k0,k0+1}, lanes>=16 hold K{k0+2,k0+3}
//   B 4x16 : lane n=L&15; same K-pair split (B[k][n] = W[n][k], contiguous in W row)
//   C/D    : vgpr r -> row r + 8*(L>=16), col L&15
// Self/neigh weights are interleaved per row in LDS (512B apart) so each n-tile's
// two B fragments pair into one ds_load_2addr_b64; fragments are preloaded in a
// batch so multiple LDS loads stay in flight instead of load->wait->wmma chains.
__launch_bounds__(256)
__global__ void sage_gemm_kernel(float* __restrict__ h, const float* __restrict__ acc,
                                 const float* __restrict__ inv_deg,
                                 const float* __restrict__ Wself,
                                 const float* __restrict__ Wneigh,
                                 const float* __restrict__ bias, int nrows) {
  extern __shared__ float lds[];
  float* wc = lds;                              // [128][260]: self|neigh interleaved
  float* bb = lds + FEAT_D * WC_PITCH;          // [128] bias
  float* ot = lds;                              // [128][132] output staging (reuses wc)

  const int tid = threadIdx.x;

  // Stage weights: row n = [Wself[n][0..127] | Wneigh[n][0..127]] at pitch 260.
  {
    int row = tid >> 1;                 // 0..127
    int cb  = (tid & 1) * 64;           // 0 or 64
    const float4* srcs = (const float4*)(Wself  + row * FEAT_D + cb);
    const float4* srcn = (const float4*)(Wneigh + row * FEAT_D + cb);
    float* ds = wc + row * WC_PITCH + cb;
    float* dn = ds + FEAT_D;
#pragma unroll
    for (int j = 0; j < 16; ++j) {
      *(float4*)(ds + j * 4) = srcs[j];
      *(float4*)(dn + j * 4) = srcn[j];
    }
    if (tid < FEAT_D) bb[tid] = bias[tid];
  }
  __syncthreads();

  const int wave = tid >> 5;
  const int lane = tid & 31;
  const int m    = lane & 15;
  const int kb   = (lane >> 4) << 1;    // 0 or 2
  const long row  = (long)blockIdx.x * 128 + wave * 16 + m;
  const long rowc = row < nrows ? row : (long)nrows - 1;   // clamp: branch-free k-loop
  const long rbase = rowc * FEAT_D;
  const float idg = inv_deg[rowc];
  const float* wbase = wc + m * WC_PITCH + kb;  // lane's B base (n = nt*16 + m)

  const v8f z8 = {0.f, 0.f, 0.f, 0.f, 0.f, 0.f, 0.f, 0.f};
  v8f c[8];
#pragma unroll
  for (int nt = 0; nt < 8; ++nt) c[nt] = z8;

  for (int k0 = 0; k0 < FEAT_D; k0 += 4) {
    v2f a_s = *(const v2f*)(h + rbase + k0 + kb);
    v2f a_n = (*(const v2f*)(acc + rbase + k0 + kb)) * idg;

    v2f bs[8], bn[8];
#pragma unroll
    for (int nt = 0; nt < 8; ++nt) {
      const float* p = wbase + nt * (16 * WC_PITCH) + k0;
      bs[nt] = *(const v2f*)(p);             // pairs with the +512B neigh fragment:
      bn[nt] = *(const v2f*)(p + FEAT_D);    // -> ds_load_2addr_b64
    }
#pragma unroll
    for (int nt = 0; nt < 8; ++nt) {
      c[nt] = __builtin_amdgcn_wmma_f32_16x16x4_f32(false, a_s, false, bs[nt],
                                                    (short)0, c[nt], false, false);
      c[nt] = __builtin_amdgcn_wmma_f32_16x16x4_f32(false, a_n, false, bn[nt],
                                                    (short)0, c[nt], false, false);
    }
  }

  // Stage bias+relu'd tile through LDS (weights no longer needed) for coalesced stores.
  __syncthreads();                       // all waves done reading wc
  {
    const int mm = (lane >> 4) * 8;
#pragma unroll
    for (int nt = 0; nt < 8; ++nt) {
      const int n = nt * 16 + (lane & 15);
      const float bv = bb[n];
#pragma unroll
      for (int r = 0; r < 8; ++r) {
        float v = c[nt][r] + bv;
        ot[(wave * 16 + mm + r) * OUT_PITCH + n] = v > 0.0f ? v : 0.0f;
      }
    }
  }
  __syncthreads();

  // Writeback: one wave per row -> 32 lanes x float4 = fully coalesced 512B row.
  // Row guard is wave-uniform (scalar branch, no per-lane predication).
  for (int r = wave; r < 128; r += 8) {
    long orow = (long)blockIdx.x * 128 + r;
    if (orow < nrows) {
      float4 v = *(const float4*)(ot + r * OUT_PITCH + lane * 4);
      *(float4*)(h + orow * FEAT_D + lane * 4) = v;
    }
  }
}

// ---------------------------------------------------------------- host launcher
extern "C" void kernel_launch(void* const* d_in, const int* in_sizes, int n_in,
                              void* d_out, int out_size, void* d_ws, size_t ws_size,
                              hipStream_t stream) {
  const float* x_src       = (const float*)d_in[0];
  const float* x_dst       = (const float*)d_in[1];
  const int*   e_src       = (const int*)d_in[2];
  const int*   e_dst       = (const int*)d_in[3];
  const float* W_ship_self = (const float*)d_in[4];
  const float* W_ship_nei  = (const float*)d_in[5];
  const float* b_ship      = (const float*)d_in[6];
  const float* W_rev_self  = (const float*)d_in[7];
  const float* W_rev_nei   = (const float*)d_in[8];
  const float* b_rev       = (const float*)d_in[9];

  const int n_src = in_sizes[0] / FEAT_D;             // 100000
  const int n_dst = in_sizes[1] / FEAT_D;             // 100000
  const int E     = in_sizes[2];                      // 500000
  const int L     = in_sizes[4] / (FEAT_D * FEAT_D);  // 2

  // h_s / h_d live directly in d_out (in-place layer updates are row-local).
  float* h_s = (float*)d_out;
  float* h_d = (float*)d_out + (size_t)n_src * FEAT_D;

  char* wsb = (char*)d_ws;
  float*    acc_d = (float*)wsb;    wsb += (size_t)n_dst * FEAT_D * sizeof(float);
  float*    acc_s = (float*)wsb;    wsb += (size_t)n_src * FEAT_D * sizeof(float);
  unsigned* deg_d = (unsigned*)wsb; wsb += (size_t)n_dst * sizeof(unsigned);
  unsigned* deg_s = (unsigned*)wsb; wsb += (size_t)n_src * sizeof(unsigned);
  float*    inv_d = (float*)wsb;    wsb += (size_t)n_dst * sizeof(float);
  float*    inv_s = (float*)wsb;

  // init h from inputs (graph-capture-safe D2D copy)
  hipMemcpyAsync(h_s, x_src, (size_t)n_src * FEAT_D * sizeof(float),
                 hipMemcpyDeviceToDevice, stream);
  hipMemcpyAsync(h_d, x_dst, (size_t)n_dst * FEAT_D * sizeof(float),
                 hipMemcpyDeviceToDevice, stream);

  // degrees (edge-static): zero, count, reciprocal
  {
    long n4 = ((long)(n_dst + n_src)) / 4;            // deg_d and deg_s are contiguous
    zero4_kernel<<<2048, 256, 0, stream>>>((float4*)deg_d, n4);
    degree_kernel<<<(E + 255) / 256, 256, 0, stream>>>(e_src, e_dst, deg_s, deg_d, E);
    invdeg_kernel<<<(n_dst + 255) / 256, 256, 0, stream>>>(deg_d, inv_d, n_dst);
    invdeg_kernel<<<(n_src + 255) / 256, 256, 0, stream>>>(deg_s, inv_s, n_src);
  }

  const size_t lds_bytes = (size_t)(FEAT_D * WC_PITCH + FEAT_D) * sizeof(float);
  const int scatter_blocks = (E + 7) / 8;             // 8 waves/block, wave per edge
  const int gemm_blocks_d  = (n_dst + 127) / 128;
  const int gemm_blocks_s  = (n_src + 127) / 128;

  for (int l = 0; l < L; ++l) {
    // zero accumulators (acc_d and acc_s are contiguous)
    zero4_kernel<<<4096, 256, 0, stream>>>((float4*)acc_d,
                                           ((long)(n_dst + n_src) * FEAT_D) / 4);
    // mean-message numerators for both directions (reads pre-update h)
    scatter_kernel<<<scatter_blocks, 256, 0, stream>>>(h_s, e_src, e_dst, acc_d, E);
    scatter_kernel<<<scatter_blocks, 256, 0, stream>>>(h_d, e_dst, e_src, acc_s, E);
    // projections + bias + relu, in place
    sage_gemm_kernel<<<gemm_blocks_d, 256, lds_bytes, stream>>>(
        h_d, acc_d, inv_d,
        W_ship_self + (size_t)l * FEAT_D * FEAT_D,
        W_ship_nei  + (size_t)l * FEAT_D * FEAT_D,
        b_ship      + (size_t)l * FEAT_D, n_dst);
    sage_gemm_kernel<<<gemm_blocks_s, 256, lds_bytes, stream>>>(
        h_s, acc_s, inv_s,
        W_rev_self + (size_t)l * FEAT_D * FEAT_D,
        W_rev_nei  + (size_t)l * FEAT_D * FEAT_D,
        b_rev      + (size_t)l * FEAT_D, n_src);
  }
}